// PhysioNetGeoLIF_4Class_6279242187141
// MI455X (gfx1250) — compile-verified
//
#include <hip/hip_runtime.h>
#include <hip/hip_bf16.h>

// PhysioNet GeoLIF, 4-class. B=4096, T=1000, C=64, K=4, D_TDA=50.
// Bandwidth-bound streaming problem (1.05 GB in @ 23.3 TB/s ~= 45us floor).
// Strategy:
//  - Fold W_spatial^T @ W_lateral into Weff[64,4] once per wave (registers).
//  - Each wave owns 16 batch rows; per time step, 16 chained
//    V_WMMA_F32_16X16X4_F32 ops contract over C=64 (full f32 precision).
//  - LIF scan state lives directly in the WMMA C/D register layout.
//  - Time axis split into 4 segments with 64-step warm-up (leak^64 ~ 6e-7)
//    to get 1024 independent waves for parallelism.
//  - Deep prefetch (12 steps = 48KB/wave) via global_prefetch_b8, which does
//    NOT consume LOADcnt, so demand b64 loads hit L2/WGP$ at short latency.

typedef __attribute__((ext_vector_type(2))) float v2f;
typedef __attribute__((ext_vector_type(8))) float v8f;

#define B_TOT 4096
#define T_TOT 1000
#define C_IN  64
#define K_OUT 4
#define D_TDA 50
#define LEAK  0.8f
#define SEG   4      // time segments
#define TS    250    // T_TOT / SEG
#define WARM  64     // warm-up steps for segments > 0
#define PFD   12     // prefetch distance in time steps

__global__ __launch_bounds__(128) void
geolif_kernel(const float* __restrict__ kin,   // [B,T,C]
              const float* __restrict__ tda,   // [B,50]
              const float* __restrict__ Wsp,   // [4,64]
              const float* __restrict__ Wlat,  // [4,4]
              const float* __restrict__ Wtda,  // [4,50]
              const float* __restrict__ btda,  // [4]
              float* __restrict__ out)         // [B,T,4]
{
    const int lane   = threadIdx.x & 31;
    const int laneLo = lane & 15;
    const int hi     = lane >> 4;               // 0: M/K low half, 1: high half
    const int wid    = (blockIdx.x * (blockDim.x >> 5)) + (threadIdx.x >> 5);
    const int seg    = wid & (SEG - 1);
    const int group  = wid >> 2;                // 0..255
    const int b0     = group * 16;

    const int tOut = seg * TS;
    const int tBeg = (seg == 0) ? 0 : (tOut - WARM);
    const int tEnd = tOut + TS;                 // trip count 250 or 314: even

    // ---- Constant B-fragments: Weff[c,k] = sum_j Wsp[j,c] * Wlat[j,k] ----
    // B 4x16 layout: lane<16 -> rows K=0,1 (comp 0,1); lane>=16 -> rows K=2,3.
    float wl[4];
#pragma unroll
    for (int j = 0; j < 4; ++j)
        wl[j] = (laneLo < K_OUT) ? Wlat[j * 4 + laneLo] : 0.0f;

    v2f bfrag[16];
#pragma unroll
    for (int i = 0; i < 16; ++i) {
        const int c0 = 4 * i + 2 * hi;
        float x = 0.0f, y = 0.0f;
#pragma unroll
        for (int j = 0; j < 4; ++j) {
            x += Wsp[j * C_IN + c0]     * wl[j];
            y += Wsp[j * C_IN + c0 + 1] * wl[j];
        }
        bfrag[i][0] = x;
        bfrag[i][1] = y;
    }

    // ---- Dynamic threshold in C/D layout: reg r <-> (M = r + 8*hi, k = laneLo)
    const int kk = (laneLo < K_OUT) ? laneLo : 0;   // dummy for unused lanes
    float thr[8];
#pragma unroll
    for (int r = 0; r < 8; ++r) {
        const int brow = b0 + r + 8 * hi;
        float acc = btda[kk];
        for (int d = 0; d < D_TDA; ++d)
            acc += tda[(size_t)brow * D_TDA + d] * Wtda[kk * D_TDA + d];
        thr[r] = 1.0f + 0.5f / (1.0f + __expf(-acc));   // base + 0.5*sigmoid
    }

    // ---- Streaming LIF loop ----
    // A 16x4 layout: lane row M = laneLo, components = channels 4i+2*hi(+1).
    const float* pfBase  =
        kin + (size_t)(b0 + laneLo) * T_TOT * C_IN;     // row start (prefetch)
    const float* rowBase = pfBase + 2 * hi;             // demand-load base

    v8f mem = {};
    v2f aCur[16], aNxt[16];

    for (int t = tBeg; t < tEnd; t += 2) {
        // Deep L2 prefetch, PFD steps ahead (clamped in-segment so addresses
        // stay inside this wave's own rows). 4 instrs cover 2 steps x 16 rows
        // x 4 cachelines: lane supplies row=laneLo, line pair selected by hi.
        {
            int tp = t + PFD;
            if (tp > tEnd - 2) tp = tEnd - 2;
            const float* pf = pfBase + (size_t)tp * C_IN + hi * 32;
            __builtin_prefetch(pf,             0, 2);
            __builtin_prefetch(pf + 16,        0, 2);
            __builtin_prefetch(pf + C_IN,      0, 2);
            __builtin_prefetch(pf + C_IN + 16, 0, 2);
        }

        // Issue both steps' loads up front (32 b64 loads in flight, NT hint).
        const float* p0 = rowBase + (size_t)t * C_IN;
        const float* p1 = p0 + C_IN;
#pragma unroll
        for (int i = 0; i < 16; ++i)
            aCur[i] = __builtin_nontemporal_load((const v2f*)(p0 + 4 * i));
#pragma unroll
        for (int i = 0; i < 16; ++i)
            aNxt[i] = __builtin_nontemporal_load((const v2f*)(p1 + 4 * i));

#pragma unroll
        for (int step = 0; step < 2; ++step) {
            const v2f* a = step ? aNxt : aCur;
            const int tc = t + step;

            v8f inh = {};
#pragma unroll
            for (int i = 0; i < 16; ++i)
                inh = __builtin_amdgcn_wmma_f32_16x16x4_f32(
                    false, a[i], false, bfrag[i], (short)0, inh, false, false);

            mem = mem * LEAK + inh;

            float sp[8];
#pragma unroll
            for (int r = 0; r < 8; ++r) {
                sp[r]  = (mem[r] >= thr[r]) ? 1.0f : 0.0f;
                mem[r] = (sp[r] != 0.0f) ? 0.0f : mem[r];   // hard reset
            }

            if (tc >= tOut) {                    // past warm-up: emit spikes
                if (laneLo < K_OUT) {            // EXEC-masked store region
                    const size_t o =
                        ((size_t)(b0 + 8 * hi) * T_TOT + tc) * K_OUT + laneLo;
#pragma unroll
                    for (int r = 0; r < 8; ++r)
                        __builtin_nontemporal_store(
                            sp[r], out + o + (size_t)r * T_TOT * K_OUT);
                }
            }
        }
    }
}

extern "C" void kernel_launch(void* const* d_in, const int* in_sizes, int n_in,
                              void* d_out, int out_size, void* d_ws, size_t ws_size,
                              hipStream_t stream) {
    const float* kin  = (const float*)d_in[0];
    const float* tda  = (const float*)d_in[1];
    const float* Wsp  = (const float*)d_in[2];
    const float* Wlat = (const float*)d_in[3];
    const float* Wtda = (const float*)d_in[4];
    const float* btda = (const float*)d_in[5];
    float* out = (float*)d_out;

    // 1024 waves total = (4096/16 batch groups) * 4 time segments.
    // 128 threads = 4 waves per block -> 256 blocks.
    dim3 grid(256), block(128);
    hipLaunchKernelGGL(geolif_kernel, grid, block, 0, stream,
                       kin, tda, Wsp, Wlat, Wtda, btda, out);
}